// SimpleAttention_5282809774659
// MI455X (gfx1250) — compile-verified
//
#include <hip/hip_runtime.h>

typedef __attribute__((ext_vector_type(2))) float v2f;
typedef __attribute__((ext_vector_type(8))) float v8f;

namespace {
constexpr int kDim    = 1024;
constexpr int kQkv    = 3072;
constexpr int kHeads  = 16;
constexpr int kHd     = 64;
constexpr int kTokens = 16384;            // B*N
constexpr int kBM     = 16;               // tokens per workgroup

// Padded LDS strides (in floats) to avoid bank conflicts (64 banks x 4B).
constexpr int kXS    = 1028;              // x / out tile row stride (1028 % 64 == 4)
constexpr int kQRow  = 68;                // per-head row stride inside q|k|v
constexpr int kQPart = kHeads * kQRow;    // 1088 floats per q|k|v part
constexpr int kQTok  = 3 * kQPart + 4;    // 3268 floats per token (skewed)
constexpr int kARow  = 18;                // attn 16x16 scratch row stride (8B-aligned rows)
constexpr int kASize = 16 * kARow;        // 288 floats per wave
constexpr int kSmemFloats = kBM * kXS + kBM * kQTok + 8 * kASize;  // 71040 -> 284160 B
}

__device__ __forceinline__ v8f wmma4(v2f a, v2f b, v8f c) {
  // D = A(16x4 f32) * B(4x16 f32) + C(16x16 f32)
  return __builtin_amdgcn_wmma_f32_16x16x4_f32(false, a, false, b, (short)0, c,
                                               false, false);
}

__device__ __forceinline__ void async_wait0() {
#if __has_builtin(__builtin_amdgcn_s_wait_asynccnt)
  __builtin_amdgcn_s_wait_asynccnt(0);
#else
  asm volatile("s_wait_asynccnt 0x0" ::: "memory");
#endif
}

__global__ __launch_bounds__(256, 1)
void fused_attn(const float* __restrict__ x,
                const float* __restrict__ wqkv,
                const float* __restrict__ bqkv,
                const float* __restrict__ wproj,
                const float* __restrict__ bproj,
                float* __restrict__ out) {
  extern __shared__ float smem[];
  float* xs = smem;                             // [16][kXS]; reused as out tile
  float* qs = smem + kBM * kXS;                 // [16][kQTok]
  float* as = smem + kBM * kXS + kBM * kQTok;   // [8][kASize]

  const int tid  = threadIdx.x;
  const int wave = tid >> 5;       // 8 waves of 32 (wave32)
  const int lane = tid & 31;
  const int hl   = lane & 15;      // position within half-wave
  const int kh   = lane >> 4;      // which K-half this lane feeds (A/B layout)
  const int row0 = blockIdx.x * kBM;

  // ---- Phase 1: async-copy x tile (16 x 1024 f32) straight into LDS ----
  // GLOBAL_LOAD_ASYNC_TO_LDS_B128, GVS mode: SGPR64 base + per-lane byte
  // offset; VDST VGPR carries the per-lane LDS byte address. Tracked by
  // ASYNCcnt (no VGPR round-trip, no LOADcnt/DScnt pressure).
  {
    const unsigned long long gbase =
        (unsigned long long)(x + (size_t)row0 * kDim);
    const unsigned int lds0 = (unsigned int)(uintptr_t)xs;  // addr[31:0] = LDS offset
    #pragma unroll
    for (int i = 0; i < (kBM * kDim / 4) / 256; ++i) {
      const int li = tid + i * 256;          // float4 index within tile
      const int r  = li >> 8;                // 256 float4 per row
      const int c4 = li & 255;
      const unsigned int ldsb = lds0 + (unsigned int)(r * kXS + c4 * 4) * 4u;
      const unsigned int goff = (unsigned int)li * 16u;  // byte offset
      asm volatile("global_load_async_to_lds_b128 %0, %1, %2"
                   :
                   : "v"(ldsb), "v"(goff), "s"(gbase)
                   : "memory");
    }
    async_wait0();
  }
  __syncthreads();

  // ---- Phase 2: QKV GEMM: qs[16][3072(padded)] = xs @ wqkv + bqkv ----
  {
    const int colW = wave * (kQkv / 8);        // 384 columns per wave
    for (int g = 0; g < 6; ++g) {
      const int colG = colW + g * 64;
      v8f acc[4] = {{}, {}, {}, {}};
      for (int k0 = 0; k0 < kDim; k0 += 4) {
        const int kk = k0 + kh * 2;
        v2f a = *(const v2f*)(xs + hl * kXS + kk);
        const float* wp = wqkv + (size_t)kk * kQkv + colG + hl;
        #pragma unroll
        for (int t = 0; t < 4; ++t) {
          v2f b;
          b.x = wp[t * 16];
          b.y = wp[kQkv + t * 16];
          acc[t] = wmma4(a, b, acc[t]);
        }
      }
      #pragma unroll
      for (int t = 0; t < 4; ++t) {
        const int col  = colG + t * 16 + hl;                 // true column
        const int cp   = (col >> 6) * kQRow + (col & 63);    // padded column
        const float bv = bqkv[col];
        #pragma unroll
        for (int r = 0; r < 8; ++r) {
          qs[(r + kh * 8) * kQTok + cp] = acc[t][r] + bv;    // row = token-in-block
        }
      }
    }
  }
  __syncthreads();

  // ---- Phase 3: per-token attention over heads (16x16 softmax) ----
  float* os = xs;  // out tile [16][kXS] reuses x-tile storage (flat 1024/row)
  for (int t2 = 0; t2 < 2; ++t2) {
    const int m = wave * 2 + t2;               // token handled by this wave
    const float* qrow = qs + m * kQTok;
    const float* krow = qrow + kQPart;
    const float* vrow = qrow + 2 * kQPart;
    float* aw = as + wave * kASize;

    // scores = q @ k^T  (16x16, K = 64)
    v8f s = {};
    #pragma unroll
    for (int k0 = 0; k0 < kHd; k0 += 4) {
      const int kk = k0 + kh * 2;
      v2f a = *(const v2f*)(qrow + hl * kQRow + kk);
      v2f b = *(const v2f*)(krow + hl * kQRow + kk);
      s = wmma4(a, b, s);
    }
    // softmax across N-axis (lanes of a 16-lane half) for each of 8 rows
    #pragma unroll
    for (int r = 0; r < 8; ++r) {
      float v = s[r] * 0.125f;                 // 1/sqrt(64)
      float mx = v;
      mx = fmaxf(mx, __shfl_xor(mx, 1));
      mx = fmaxf(mx, __shfl_xor(mx, 2));
      mx = fmaxf(mx, __shfl_xor(mx, 4));
      mx = fmaxf(mx, __shfl_xor(mx, 8));
      float e = __expf(v - mx);
      float sm = e;
      sm += __shfl_xor(sm, 1);
      sm += __shfl_xor(sm, 2);
      sm += __shfl_xor(sm, 4);
      sm += __shfl_xor(sm, 8);
#if __has_builtin(__builtin_amdgcn_rcpf)
      aw[(r + kh * 8) * kARow + hl] = e * __builtin_amdgcn_rcpf(sm);
#else
      aw[(r + kh * 8) * kARow + hl] = e / sm;
#endif
    }
    __syncthreads();   // uniform; also orders the C->A layout bounce via LDS

    // out = attn @ v  (16x64, K = 16)
    v8f o[4] = {{}, {}, {}, {}};
    #pragma unroll
    for (int k0 = 0; k0 < 16; k0 += 4) {
      const int kk = k0 + kh * 2;
      v2f a = *(const v2f*)(aw + hl * kARow + kk);
      #pragma unroll
      for (int t = 0; t < 4; ++t) {
        v2f b;
        b.x = vrow[kk * kQRow + t * 16 + hl];
        b.y = vrow[(kk + 1) * kQRow + t * 16 + hl];
        o[t] = wmma4(a, b, o[t]);
      }
    }
    float* orow = os + m * kXS;                // flat [h*64 + d] within row
    #pragma unroll
    for (int t = 0; t < 4; ++t) {
      #pragma unroll
      for (int r = 0; r < 8; ++r) {
        orow[(r + kh * 8) * kHd + t * 16 + hl] = o[t][r];
      }
    }
  }
  __syncthreads();

  // ---- Phase 4: proj GEMM: out[16][1024] = os @ wproj + bproj ----
  for (int g = 0; g < 2; ++g) {
    const int colG = wave * 128 + g * 64;
    v8f acc[4] = {{}, {}, {}, {}};
    for (int k0 = 0; k0 < kDim; k0 += 4) {
      const int kk = k0 + kh * 2;
      v2f a = *(const v2f*)(os + hl * kXS + kk);
      const float* wp = wproj + (size_t)kk * kDim + colG + hl;
      #pragma unroll
      for (int t = 0; t < 4; ++t) {
        v2f b;
        b.x = wp[t * 16];
        b.y = wp[kDim + t * 16];
        acc[t] = wmma4(a, b, acc[t]);
      }
    }
    #pragma unroll
    for (int t = 0; t < 4; ++t) {
      const int col  = colG + t * 16 + hl;
      const float bv = bproj[col];
      #pragma unroll
      for (int r = 0; r < 8; ++r) {
        out[(size_t)(row0 + r + kh * 8) * kDim + col] = acc[t][r] + bv;
      }
    }
  }
}

extern "C" void kernel_launch(void* const* d_in, const int* in_sizes, int n_in,
                              void* d_out, int out_size, void* d_ws, size_t ws_size,
                              hipStream_t stream) {
  (void)in_sizes; (void)n_in; (void)out_size; (void)d_ws; (void)ws_size;
  const float* x     = (const float*)d_in[0];
  const float* wqkv  = (const float*)d_in[1];
  const float* bqkv  = (const float*)d_in[2];
  const float* wproj = (const float*)d_in[3];
  const float* bproj = (const float*)d_in[4];
  float* out = (float*)d_out;

  const size_t shmem = (size_t)kSmemFloats * sizeof(float);  // ~278 KB of 320 KB/WGP
  (void)hipFuncSetAttribute((const void*)fused_attn,
                            hipFuncAttributeMaxDynamicSharedMemorySize,
                            (int)shmem);
  fused_attn<<<dim3(kTokens / kBM), dim3(256), shmem, stream>>>(
      x, wqkv, bqkv, wproj, bproj, out);
}